// MTGP_64175401337007
// MI455X (gfx1250) — compile-verified
//
#include <hip/hip_runtime.h>
#include <hip/hip_bf16.h>

typedef __attribute__((ext_vector_type(2))) float v2f;
typedef __attribute__((ext_vector_type(8))) float v8f;

#define M_TOT   4096
#define D_DIM   64
#define Z_LAT   4
#define JITTER  0.002f

// ---------------------------------------------------------------------------
// Pass 1: per-row squared norms sq[i] = ||X[i]||^2  (4096 rows x 64 floats)
// ---------------------------------------------------------------------------
__global__ __launch_bounds__(256) void mtgp_rowsq_kernel(
    const float* __restrict__ X, float* __restrict__ sq)
{
    const int row = blockIdx.x * blockDim.x + threadIdx.x;   // 4096 threads
    const float4* p = (const float4*)(X + (size_t)row * D_DIM);
    float s = 0.0f;
#pragma unroll
    for (int k = 0; k < D_DIM / 4; ++k) {
        float4 v = p[k];
        s = fmaf(v.x, v.x, s);
        s = fmaf(v.y, v.y, s);
        s = fmaf(v.z, v.z, s);
        s = fmaf(v.w, v.w, s);
    }
    sq[row] = s;
}

// ---------------------------------------------------------------------------
// Pass 2: one 16x16 covariance tile per wave.
// Gram tile via 16 chained V_WMMA_F32_16X16X4_F32 (exact fp32), then the
// Z=4 latent-GP RBF mix + diagonal jitter, fused in registers.
//
// Fragment layout (CDNA5 ISA 7.12.2, 32-bit A 16x4 / B 4x16 / C 16x16):
//   h = lane>=16, r = lane&15
//   A vgpr v : X[i0 + r][k0 + 2h + v]      (v = 0,1)
//   B vgpr v : X[j0 + r][k0 + 2h + v]      (B = Xj^T in K-major striping)
//   C elem e : row i0 + 8h + e, col j0 + r
// ---------------------------------------------------------------------------
__global__ __launch_bounds__(256) void mtgp_cov_kernel(
    const float* __restrict__ X,
    const float* __restrict__ W,       // [T=4, Z=4]
    const float* __restrict__ theta,   // [Z=4, 2]
    const float* __restrict__ sq,      // [4096]
    float* __restrict__ out)           // [4096, 4096]
{
    const int lane = threadIdx.x & 31;
    const int wave = threadIdx.x >> 5;
    const int tile = blockIdx.x * 8 + wave;        // 65536 tiles total
    const int ti = tile >> 8;                      // 256 tiles per dim
    const int tj = tile & 255;
    const int i0 = ti << 4;
    const int j0 = tj << 4;

    const int h = lane >> 4;                       // lane half
    const int r = lane & 15;

    const float* Arow = X + (size_t)(i0 + r) * D_DIM + (h << 1);
    const float* Brow = X + (size_t)(j0 + r) * D_DIM + (h << 1);

    v8f c = {0.f, 0.f, 0.f, 0.f, 0.f, 0.f, 0.f, 0.f};
#pragma unroll
    for (int k = 0; k < D_DIM / 4; ++k) {
        v2f a = *(const v2f*)(Arow + (k << 2));
        v2f b = *(const v2f*)(Brow + (k << 2));
        // D = A x B + C, fp32 throughout
        c = __builtin_amdgcn_wmma_f32_16x16x4_f32(
                /*neg_a=*/false, a, /*neg_b=*/false, b,
                /*c_mod=*/(short)0, c, /*reuse_a=*/false, /*reuse_b=*/false);
    }

    // Column handled by this lane; tasks are uniform per 16-wide tile (16 | 1024).
    const int   j     = j0 + r;
    const float sqj   = sq[j];
    const int   taskj = j0 >> 10;
    const int   taski = i0 >> 10;

    float amp[Z_LAT], nhl[Z_LAT], wij[Z_LAT];
#pragma unroll
    for (int z = 0; z < Z_LAT; ++z) {
        amp[z] = theta[z * 2 + 0];
        nhl[z] = -0.5f * theta[z * 2 + 1];
        wij[z] = W[taski * Z_LAT + z] * W[taskj * Z_LAT + z] * amp[z];
    }

#pragma unroll
    for (int e = 0; e < 8; ++e) {
        const int   i   = i0 + (h << 3) + e;
        float sqd = sq[i] + sqj - 2.0f * c[e];
        sqd = fmaxf(sqd, 0.0f);
        float acc = (i == j) ? JITTER : 0.0f;
#pragma unroll
        for (int z = 0; z < Z_LAT; ++z)
            acc = fmaf(wij[z], __expf(nhl[z] * sqd), acc);   // v_exp_f32 path
        out[(size_t)i * M_TOT + j] = acc;
    }
}

// ---------------------------------------------------------------------------
extern "C" void kernel_launch(void* const* d_in, const int* in_sizes, int n_in,
                              void* d_out, int out_size, void* d_ws, size_t ws_size,
                              hipStream_t stream)
{
    const float* x     = (const float*)d_in[0];   // [4,1024,64] fp32
    const float* W     = (const float*)d_in[1];   // [4,4] fp32
    const float* theta = (const float*)d_in[2];   // [4,2] fp32
    float* out = (float*)d_out;                   // [4096,4096] fp32
    float* sq  = (float*)d_ws;                    // 4096 floats scratch

    mtgp_rowsq_kernel<<<M_TOT / 256, 256, 0, stream>>>(x, sq);
    mtgp_cov_kernel<<<(256 * 256) / 8, 256, 0, stream>>>(x, W, theta, sq, out);
}